// LocalMean_8821862826420
// MI455X (gfx1250) — compile-verified
//
#include <hip/hip_runtime.h>

// 5x5 box-filter mean with reflect padding over (32, 3, 512, 512) f32.
// Memory-bound: ~201 MB min traffic -> ~8.6 us floor at 23.3 TB/s.
// - Full-row 32-row tiles: no horizontal HBM halo, vertical amplification 1.125x.
// - Async global->LDS staging (GLOBAL_LOAD_ASYNC_TO_LDS_B128 + s_wait_asynccnt).
// - Separable filter: each thread owns 2 adjacent columns; the shared 6-float
//   window is read as 3x ds_load_b64 (8B-aligned, bank-conflict-free) and the
//   pair of outputs is written as one coalesced float2 store.

#define PAD_      2
#define N_        32
#define C_        3
#define H_        512
#define W_        512
#define TILE_H    32
#define LROWS     (TILE_H + 2 * PAD_)   // 36 staged rows
#define XOFF      4                     // x=0 lives at float index 4 (16B aligned)
#define LSTRIDE   520                   // floats per LDS row (2..517 used), 2080B pitch

#define AS1 __attribute__((address_space(1)))
#define AS3 __attribute__((address_space(3)))

typedef int v4i __attribute__((ext_vector_type(4)));

#if defined(__gfx1250__) && __has_builtin(__builtin_amdgcn_global_load_async_to_lds_b128)
#define USE_ASYNC_LDS 1
#else
#define USE_ASYNC_LDS 0
#endif

__global__ __launch_bounds__(256)
void LocalMean_box5_kernel(const float* __restrict__ in, float* __restrict__ out) {
    extern __shared__ float tile[];   // LROWS * LSTRIDE floats (74,880 B)

    const int tid   = threadIdx.x;
    const int plane = blockIdx.y;                 // n*C + c
    const int row0  = blockIdx.x * TILE_H;        // first output row of this tile
    const float* __restrict__ src = in  + (size_t)plane * (H_ * W_);
    float*       __restrict__ dst = out + (size_t)plane * (H_ * W_);

    // ---- Stage LROWS x 512 floats into LDS (vertical reflect at the seams) ----
    // 36 rows * 128 float4 = 4608 chunks; 256 threads -> 18 iterations each.
    for (int i = tid; i < LROWS * (W_ / 4); i += 256) {
        const int r = i >> 7;          // row within tile (0..35)
        const int v = i & 127;         // float4 index within row
        int g = row0 - PAD_ + r;       // global source row, reflect-mode
        g = (g < 0) ? -g : g;
        g = (g >= H_) ? (2 * H_ - 2 - g) : g;
        const float* gp = src + (size_t)g * W_ + v * 4;
        float*       lp = &tile[r * LSTRIDE + XOFF + v * 4];
#if USE_ASYNC_LDS
        __builtin_amdgcn_global_load_async_to_lds_b128(
            (AS1 v4i*)gp, (AS3 v4i*)lp, 0, 0);
#else
        *(float4*)lp = *(const float4*)gp;
#endif
    }
#if USE_ASYNC_LDS
#if __has_builtin(__builtin_amdgcn_s_wait_asynccnt)
    __builtin_amdgcn_s_wait_asynccnt(0);
#else
    asm volatile("s_wait_asynccnt 0" ::: "memory");
#endif
#endif
    __syncthreads();

    // ---- Horizontal reflect halo: 4 elements per row, copied inside LDS ----
    if (tid < LROWS * 4) {
        const int r = tid >> 2;
        const int k = tid & 3;
        // (dst x, src x) pairs: (-2,2) (-1,1) (512,510) (513,509)
        const int dx = (k == 0) ? -2 : (k == 1) ? -1 : (k == 2) ? 512 : 513;
        const int sx = (k == 0) ?  2 : (k == 1) ?  1 : (k == 2) ? 510 : 509;
        tile[r * LSTRIDE + XOFF + dx] = tile[r * LSTRIDE + XOFF + sx];
    }
    __syncthreads();

    // ---- Separable 5x5: two adjacent columns per thread ----
    // Window for outputs x=2t, 2t+1 spans floats [2t-2 .. 2t+3] ->
    // LDS float index (XOFF + 2t - 2) = 2 + 2t : even -> three 8B-aligned b64 loads.
    const float inv25 = 1.0f / 25.0f;
    const int   xb    = 2 * tid;                  // first of the two output columns
    const int   cbase = 2 + xb;                   // LDS float index of window start

    // hpair(r): {ha, hb} horizontal 5-sums for the two columns at staged row r.
#define HPAIR(r, ha, hb) do {                                            \
        const float2 p0 = *(const float2*)&tile[(r) * LSTRIDE + cbase];     \
        const float2 p1 = *(const float2*)&tile[(r) * LSTRIDE + cbase + 2]; \
        const float2 p2 = *(const float2*)&tile[(r) * LSTRIDE + cbase + 4]; \
        const float  m  = (p0.y + p1.x) + (p1.y + p2.x);                    \
        (ha) = m + p0.x;                                                    \
        (hb) = m + p2.y;                                                    \
    } while (0)

    float a0, a1, a2, a3, b0, b1, b2, b3;
    HPAIR(0, a0, b0);
    HPAIR(1, a1, b1);
    HPAIR(2, a2, b2);
    HPAIR(3, a3, b3);

    float* __restrict__ drow = dst + (size_t)row0 * W_;
#pragma unroll
    for (int j = 0; j < TILE_H; ++j) {
        float a4, b4;
        HPAIR(j + 4, a4, b4);
        float2 o;
        o.x = ((a0 + a1) + (a2 + a3) + a4) * inv25;
        o.y = ((b0 + b1) + (b2 + b3) + b4) * inv25;
        *(float2*)&drow[j * W_ + xb] = o;
        a0 = a1; a1 = a2; a2 = a3; a3 = a4;
        b0 = b1; b1 = b2; b2 = b3; b3 = b4;
    }
#undef HPAIR
}

extern "C" void kernel_launch(void* const* d_in, const int* in_sizes, int n_in,
                              void* d_out, int out_size, void* d_ws, size_t ws_size,
                              hipStream_t stream) {
    (void)in_sizes; (void)n_in; (void)out_size; (void)d_ws; (void)ws_size;
    const float* img = (const float*)d_in[0];
    float*       out = (float*)d_out;

    dim3 grid(H_ / TILE_H, N_ * C_);           // 16 x 96 = 1536 blocks
    const size_t lds_bytes = (size_t)LROWS * LSTRIDE * sizeof(float);  // 74,880 B
    LocalMean_box5_kernel<<<grid, 256, lds_bytes, stream>>>(img, out);
}